// EpsilonNetGM_4209067950718
// MI455X (gfx1250) — compile-verified
//
#include <hip/hip_runtime.h>
#include <math.h>

typedef __attribute__((ext_vector_type(2))) float v2f;
typedef __attribute__((ext_vector_type(8))) float v8f;

#define N_PTS 32768
#define KC    25
#define DIM   128
#define WAVES_PER_BLOCK 8
#define LDS_STRIDE 34   // even (keeps float2 8B alignment), breaks half-wave bank aliasing

// ---------------------------------------------------------------------------
// Prologue: per-component constants.
//   ws[0..31]  : c_k = log(w_k) - 0.5*acp*||mu_k||^2/sigma2   (-1e30 for pad k)
//   ws[32]     : scale = sqrt(acp)/sigma2   (logit scale on G = x.mu)
//   ws[33]     : s     = sqrt(acp)
//   ws[34]     : 1/sqrt(sigma2)
// ---------------------------------------------------------------------------
__global__ void eps_gmm_prologue(const float* __restrict__ means,
                                 const float* __restrict__ weights,
                                 const float* __restrict__ acp_arr,
                                 const int*   __restrict__ t_arr,
                                 float* __restrict__ ws) {
    const int k = threadIdx.x;           // 0..31
    const int t = t_arr[0];
    const float acp    = acp_arr[t];
    const float sigma2 = 1.0f - acp;
    float c = -1e30f;
    if (k < KC) {
        float mm = 0.0f;
        for (int d = 0; d < DIM; ++d) {
            float m = means[k * DIM + d];
            mm = fmaf(m, m, mm);
        }
        c = __logf(weights[k]) - 0.5f * acp * mm / sigma2;
    }
    ws[k] = c;
    if (k == 0) {
        float s = sqrtf(acp);
        ws[32] = s / sigma2;
        ws[33] = s;
        ws[34] = rsqrtf(sigma2);
    }
}

// ---------------------------------------------------------------------------
// Main kernel: one wave32 per 16-row tile of x.
//   GEMM1  G = Xtile(16xD) * Means^T(DxKpad)  via V_WMMA_F32_16X16X4_F32
//   softmax over k (cross-lane, 16-lane halves)
//   transpose resp through per-wave LDS (C-layout -> A-frag layout)
//   GEMM2  P = R(16x32) * Means(32xD), out = (x - s*P)/sqrt(sigma2)
// ---------------------------------------------------------------------------
__global__ __launch_bounds__(256) void eps_gmm_kernel(
        const float* __restrict__ x,
        const float* __restrict__ means,
        const float* __restrict__ ws,
        float* __restrict__ out) {
    __shared__ float resp_lds[WAVES_PER_BLOCK][16 * LDS_STRIDE];

    const int lane  = threadIdx.x & 31;
    const int wave  = threadIdx.x >> 5;
    const int tile  = blockIdx.x * WAVES_PER_BLOCK + wave;
    const int row0  = tile * 16;
    const int lmod  = lane & 15;        // M (GEMM A / C col index)
    const int khalf = lane >> 4;        // which K-pair half of the wave
    const int doff  = khalf * 2;        // K-offset within a 4-wide k-step

    const float c0      = ws[lmod];           // c_k for k = lmod      (tile 0)
    const float c1      = ws[16 + lmod];      // c_k for k = 16 + lmod (tile 1)
    const float scale   = ws[32];
    const float s       = ws[33];
    const float inv_std = ws[34];

    // ---- GEMM1: preload A-fragments (this wave's 16 x-rows, fp32) ----------
    const float* xrow = x + (size_t)(row0 + lmod) * DIM;
    v2f afrag[32];
#pragma unroll
    for (int i = 0; i < 32; ++i)
        afrag[i] = *(const v2f*)(xrow + 4 * i + doff);

    // B-fragments: B[d][k] = means[k][d]  (k >= KC clamped; masked by c_k)
    const int kb1 = (16 + lmod) < KC ? (16 + lmod) : (KC - 1);
    const float* mrow0 = means + (size_t)lmod * DIM;
    const float* mrow1 = means + (size_t)kb1  * DIM;

    v8f acc0 = {}; v8f acc1 = {};
#pragma unroll
    for (int i = 0; i < 32; ++i) {
        v2f b0 = *(const v2f*)(mrow0 + 4 * i + doff);
        v2f b1 = *(const v2f*)(mrow1 + 4 * i + doff);
        acc0 = __builtin_amdgcn_wmma_f32_16x16x4_f32(false, afrag[i], false, b0,
                                                     (short)0, acc0, false, false);
        acc1 = __builtin_amdgcn_wmma_f32_16x16x4_f32(false, afrag[i], false, b1,
                                                     (short)0, acc1, false, false);
    }

    // ---- logits + softmax over k; write responsibilities to LDS ------------
    // C layout: vgpr r, lane l -> row m = r + 8*(l>=16), k = l%16 (+16 for acc1)
    float* myl = &resp_lds[wave][0];
#pragma unroll
    for (int r = 0; r < 8; ++r) {
        float l0 = fmaf(scale, acc0[r], c0);
        float l1 = fmaf(scale, acc1[r], c1);
        float mx = fmaxf(l0, l1);
#pragma unroll
        for (int m = 1; m <= 8; m <<= 1)
            mx = fmaxf(mx, __shfl_xor(mx, m, 32));   // reduce within 16-lane half
        float e0 = __expf(l0 - mx);
        float e1 = __expf(l1 - mx);
        float sum = e0 + e1;
#pragma unroll
        for (int m = 1; m <= 8; m <<= 1)
            sum += __shfl_xor(sum, m, 32);
        float inv = 1.0f / sum;
        int mrow = r + 8 * khalf;
        myl[mrow * LDS_STRIDE + lmod]      = e0 * inv;   // k = lmod
        myl[mrow * LDS_STRIDE + 16 + lmod] = e1 * inv;   // k = 16 + lmod
    }
    // Same-wave LDS ops are in-order: store->load transpose needs no barrier.

    // ---- GEMM2: P = R(16x32) @ Means(32 x 16-col blocks) --------------------
#pragma unroll
    for (int db = 0; db < 8; ++db) {
        const int n0 = db * 16;
        v8f acc = {};
#pragma unroll
        for (int j = 0; j < 8; ++j) {
            const int kc = 4 * j + doff;
            // A-frag (resp): lane l -> M = l%16, K = kc, kc+1 (float2 from LDS)
            v2f a = *(const v2f*)(myl + lmod * LDS_STRIDE + kc);
            // B-frag: B[k][n] = means[k][n0 + l%16]; clamp pad k (resp==0 there)
            const int ka = kc     < KC ? kc     : (KC - 1);
            const int kb = kc + 1 < KC ? kc + 1 : (KC - 1);
            v2f b;
            b.x = means[(size_t)ka * DIM + n0 + lmod];
            b.y = means[(size_t)kb * DIM + n0 + lmod];
            acc = __builtin_amdgcn_wmma_f32_16x16x4_f32(false, a, false, b,
                                                        (short)0, acc, false, false);
        }
        // epilogue: out = (x - s*P) / sqrt(sigma2)
#pragma unroll
        for (int r = 0; r < 8; ++r) {
            const int row = row0 + r + 8 * khalf;
            const int d   = n0 + lmod;
            const float xv = x[(size_t)row * DIM + d];
            out[(size_t)row * DIM + d] = (xv - s * acc[r]) * inv_std;
        }
    }
}

// ---------------------------------------------------------------------------
extern "C" void kernel_launch(void* const* d_in, const int* in_sizes, int n_in,
                              void* d_out, int out_size, void* d_ws, size_t ws_size,
                              hipStream_t stream) {
    const float* x       = (const float*)d_in[0];   // [N, D]
    const float* means   = (const float*)d_in[1];   // [K, D]
    const float* weights = (const float*)d_in[2];   // [K]
    const float* acp     = (const float*)d_in[3];   // [T]
    const int*   t       = (const int*)  d_in[4];   // [1]
    float* ws  = (float*)d_ws;
    float* out = (float*)d_out;

    eps_gmm_prologue<<<dim3(1), dim3(32), 0, stream>>>(means, weights, acp, t, ws);

    const int tiles  = N_PTS / 16;                  // 2048 wave-tiles
    const int blocks = tiles / WAVES_PER_BLOCK;     // 256 blocks x 8 waves
    eps_gmm_kernel<<<dim3(blocks), dim3(256), 0, stream>>>(x, means, ws, out);
}